// GCNResBlock_old_4329327034522
// MI455X (gfx1250) — compile-verified
//
#include <hip/hip_runtime.h>

typedef __attribute__((ext_vector_type(2))) float v2f;
typedef __attribute__((ext_vector_type(8))) float v8f;

#define NODES 50000
#define INCH 128
#define OUTCH 256
#define BN_EPS 1e-5f

// ---------------------------------------------------------------------------
// k1: init deg=1 (self loop) and zero the column-stat accumulators
// ---------------------------------------------------------------------------
__global__ __launch_bounds__(256) void k_init(float* __restrict__ deg,
                                              float* __restrict__ sum,
                                              float* __restrict__ sumsq,
                                              int n) {
  int i = blockIdx.x * blockDim.x + threadIdx.x;
  if (i < n) deg[i] = 1.0f;
  if (i < OUTCH) { sum[i] = 0.0f; sumsq[i] = 0.0f; }
}

// ---------------------------------------------------------------------------
// k2: deg[dst] += 1 per edge
// ---------------------------------------------------------------------------
__global__ __launch_bounds__(256) void k_deg(const int* __restrict__ dst,
                                             float* __restrict__ deg, int E) {
  int e = blockIdx.x * blockDim.x + threadIdx.x;
  if (e < E) atomicAdd(&deg[dst[e]], 1.0f);
}

// ---------------------------------------------------------------------------
// k3: deg -> rsqrt(deg) in place (deg >= 1 always, no zero check needed)
// ---------------------------------------------------------------------------
__global__ __launch_bounds__(256) void k_rsqrt(float* __restrict__ deg, int n) {
  int i = blockIdx.x * blockDim.x + threadIdx.x;
  if (i < n) deg[i] = rsqrtf(deg[i]);
}

// ---------------------------------------------------------------------------
// k4/k5: fp32 WMMA GEMM  C[M,Nc] = A[M,K] @ B[K,Nc]
// One wave per 16x16 C tile, V_WMMA_F32_16X16X4_F32, K stepped by 4.
// A layout (ISA 7.12.2, 32-bit A 16x4): lane%16 = M, lane/16 selects K pair
//   vgpr0 = K(2*half), vgpr1 = K(2*half+1)
// B layout mirrors with N = lane%16.
// C/D: vgpr r -> row r (lanes 0-15) / row r+8 (lanes 16-31), col = lane%16.
// ---------------------------------------------------------------------------
__global__ __launch_bounds__(128) void k_gemm_wmma(const float* __restrict__ A,
                                                   const float* __restrict__ B,
                                                   float* __restrict__ C,
                                                   int M, int K, int Nc) {
  const int lane = threadIdx.x & 31;
  const int wave = threadIdx.x >> 5;
  const int ntiles = Nc >> 4;
  const int tile = blockIdx.x * 4 + wave;
  const int mtile = tile / ntiles;
  const int ntile = tile - mtile * ntiles;
  if (mtile * 16 >= M) return;  // wave-uniform: EXEC stays all-1s for WMMA

  const int mn = lane & 15;                 // M (for A) / N (for B,C)
  const int khalf = (lane >> 4) * 2;        // which K pair this lane holds
  const float* Arow = A + (size_t)(mtile * 16 + mn) * K;
  const float* Bcol = B + (size_t)(ntile * 16 + mn);

  v8f acc = {};
#pragma unroll 8
  for (int kk = 0; kk < K; kk += 4) {
    v2f a, b;
    a.x = Arow[kk + khalf];
    a.y = Arow[kk + khalf + 1];
    b.x = Bcol[(size_t)(kk + khalf) * Nc];
    b.y = Bcol[(size_t)(kk + khalf + 1) * Nc];
    acc = __builtin_amdgcn_wmma_f32_16x16x4_f32(
        /*neg_a=*/false, a, /*neg_b=*/false, b,
        /*c_mod=*/(short)0, acc, /*reuse_a=*/false, /*reuse_b=*/false);
  }

  const int rowbase = mtile * 16 + ((lane >> 4) << 3);
  const int col = ntile * 16 + mn;
#pragma unroll
  for (int r = 0; r < 8; ++r)
    C[(size_t)(rowbase + r) * Nc + col] = acc[r];
}

// ---------------------------------------------------------------------------
// k6: agg[i,:] = b + h[i,:] * dinv[i]^2   (self-loop term + bias), float4
// ---------------------------------------------------------------------------
__global__ __launch_bounds__(256) void k_agg_init(const float* __restrict__ h,
                                                  const float* __restrict__ b,
                                                  const float* __restrict__ dinv,
                                                  float* __restrict__ agg,
                                                  int total4) {
  int i = blockIdx.x * blockDim.x + threadIdx.x;
  if (i >= total4) return;
  int row = i >> 6;            // 64 float4 per 256-wide row
  int c = (i & 63) << 2;
  float di = dinv[row];
  float w = di * di;
  float4 hv = ((const float4*)h)[i];
  float4 r;
  r.x = b[c + 0] + hv.x * w;
  r.y = b[c + 1] + hv.y * w;
  r.z = b[c + 2] + hv.z * w;
  r.w = b[c + 3] + hv.w * w;
  ((float4*)agg)[i] = r;
}

// ---------------------------------------------------------------------------
// k7: wave-per-edge scatter: agg[dst,:] += h[src,:] * dinv[src]*dinv[dst]
// Each lane owns 8 columns (two float4 gathers); 256 f32 atomics per edge.
// h and agg are L2-resident (51 MB each < 192 MB L2).
// ---------------------------------------------------------------------------
__global__ __launch_bounds__(256) void k_scatter(const int* __restrict__ src,
                                                 const int* __restrict__ dst,
                                                 const float* __restrict__ h,
                                                 const float* __restrict__ dinv,
                                                 float* __restrict__ agg, int E) {
  int gid = blockIdx.x * blockDim.x + threadIdx.x;
  int e = gid >> 5;
  int lane = gid & 31;
  if (e >= E) return;
  int s = src[e];
  int d = dst[e];
  float norm = dinv[s] * dinv[d];
  const float4* hs = (const float4*)(h + (size_t)s * OUTCH);
  float* ad = agg + (size_t)d * OUTCH;
#pragma unroll
  for (int q = 0; q < 2; ++q) {
    int v4 = lane + q * 32;
    float4 v = hs[v4];
    int c = v4 << 2;
    atomicAdd(ad + c + 0, v.x * norm);
    atomicAdd(ad + c + 1, v.y * norm);
    atomicAdd(ad + c + 2, v.z * norm);
    atomicAdd(ad + c + 3, v.w * norm);
  }
}

// ---------------------------------------------------------------------------
// k8: per-column sum / sum-of-squares (thread owns one column, row loop,
// register accumulation, 2 atomics per thread at the end)
// ---------------------------------------------------------------------------
__global__ __launch_bounds__(256) void k_colstats(const float* __restrict__ agg,
                                                  float* __restrict__ sum,
                                                  float* __restrict__ sumsq,
                                                  int nrows) {
  int c = threadIdx.x;
  int per = (nrows + gridDim.x - 1) / gridDim.x;
  int r0 = blockIdx.x * per;
  int r1 = r0 + per;
  if (r1 > nrows) r1 = nrows;
  float s = 0.0f, sq = 0.0f;
  for (int r = r0; r < r1; ++r) {
    float v = agg[(size_t)r * OUTCH + c];
    s += v;
    sq += v * v;
  }
  atomicAdd(&sum[c], s);
  atomicAdd(&sumsq[c], sq);
}

// ---------------------------------------------------------------------------
// k9: out = relu((agg - mean) * rsqrt(var+eps) * gamma + beta) + out
// (out already holds x @ Wp). Biased variance = E[x^2] - mean^2.
// ---------------------------------------------------------------------------
__global__ __launch_bounds__(256) void k_final(const float* __restrict__ agg,
                                               const float* __restrict__ sum,
                                               const float* __restrict__ sumsq,
                                               const float* __restrict__ gamma,
                                               const float* __restrict__ beta,
                                               float* __restrict__ out,
                                               int total4, float invN) {
  int i = blockIdx.x * blockDim.x + threadIdx.x;
  if (i >= total4) return;
  int c = (i << 2) & (OUTCH - 1);
  float4 a = ((const float4*)agg)[i];
  float4 o = ((float4*)out)[i];
  float4 r;
#pragma unroll
  for (int q = 0; q < 4; ++q) {
    float mean = sum[c + q] * invN;
    float var = sumsq[c + q] * invN - mean * mean;
    float scale = rsqrtf(var + BN_EPS) * gamma[c + q];
    float av = (&a.x)[q];
    float bn = (av - mean) * scale + beta[c + q];
    bn = bn > 0.0f ? bn : 0.0f;
    (&r.x)[q] = bn + (&o.x)[q];
  }
  ((float4*)out)[i] = r;
}

// ---------------------------------------------------------------------------
extern "C" void kernel_launch(void* const* d_in, const int* in_sizes, int n_in,
                              void* d_out, int out_size, void* d_ws, size_t ws_size,
                              hipStream_t stream) {
  const float* x     = (const float*)d_in[0];
  const int*   ei    = (const int*)d_in[1];
  const float* W     = (const float*)d_in[2];
  const float* b     = (const float*)d_in[3];
  const float* gamma = (const float*)d_in[4];
  const float* beta  = (const float*)d_in[5];
  const float* Wp    = (const float*)d_in[6];
  float* out = (float*)d_out;

  const int n = in_sizes[0] / INCH;   // 50000
  const int E = in_sizes[1] / 2;      // 800000
  const int* src = ei;
  const int* dst = ei + E;

  // workspace layout
  float* h     = (float*)d_ws;                     // n*256
  float* agg   = h + (size_t)n * OUTCH;            // n*256
  float* deg   = agg + (size_t)n * OUTCH;          // n  (becomes dinv)
  float* sum   = deg + n;                          // 256
  float* sumsq = sum + OUTCH;                      // 256

  const int total4 = n * (OUTCH / 4);

  // degrees + norms
  k_init<<<(n + 255) / 256, 256, 0, stream>>>(deg, sum, sumsq, n);
  k_deg<<<(E + 255) / 256, 256, 0, stream>>>(dst, deg, E);
  k_rsqrt<<<(n + 255) / 256, 256, 0, stream>>>(deg, n);

  // GEMMs: h = x@W (ws), proj = x@Wp (directly into d_out)
  const int tiles = (n / 16) * (OUTCH / 16);       // 50000
  const int gblocks = (tiles + 3) / 4;
  k_gemm_wmma<<<gblocks, 128, 0, stream>>>(x, W, h, n, INCH, OUTCH);
  k_gemm_wmma<<<gblocks, 128, 0, stream>>>(x, Wp, out, n, INCH, OUTCH);

  // aggregation
  k_agg_init<<<(total4 + 255) / 256, 256, 0, stream>>>(h, b, deg, agg, total4);
  k_scatter<<<(E * 32 + 255) / 256, 256, 0, stream>>>(src, dst, h, deg, agg, E);

  // batchnorm stats + fused epilogue
  k_colstats<<<200, 256, 0, stream>>>(agg, sum, sumsq, n);
  k_final<<<(total4 + 255) / 256, 256, 0, stream>>>(agg, sum, sumsq, gamma, beta,
                                                    out, total4, 1.0f / (float)n);
}